// GAT_84559316124074
// MI455X (gfx1250) — compile-verified
//
#include <hip/hip_runtime.h>
#include <hip/hip_bf16.h>
#include <math.h>

typedef __attribute__((ext_vector_type(2))) float v2f;
typedef __attribute__((ext_vector_type(8))) float v8f;

#define NEG_SLOPE 0.2f

static __device__ __forceinline__ void atomicMaxF(float* addr, float v) {
  // sign-aware float max via integer atomics (monotone bit patterns)
  if (v >= 0.f) atomicMax((int*)addr, __float_as_int(v));
  else          atomicMin((unsigned int*)addr, (unsigned int)__float_as_int(v));
}

// ---------------------------------------------------------------------------
// WMMA GEMM: C[M,Nc] = A[M,K] @ B[K,Nc]   (fp32, V_WMMA_F32_16X16X4_F32)
// One wave -> one 16-row stripe, NT 16-col accumulator tiles (compile-time).
// B staged in LDS, zero-padded, pair-interleaved so each lane's {kb,kb+1}
// fragment is one aligned ds_load_b64:
//   Bs[(k>>1)*(2*ldbs) + 2*n + (k&1)] = B[k][n]
// Fragment layout assumption (ISA 7.12.2, 32-bit A 16x4 / B 4x16):
//   A: lane L: row = L&15, regs = K {kb, kb+1}, kb = 2*(L>>4)
//   B: lane L: col = L&15, regs = K {kb, kb+1}
//   C: reg r: row = r + 8*(L>>4), col = L&15
// ---------------------------------------------------------------------------
template <int NT, int K>
__global__ __launch_bounds__(256) void k_wmma_gemm(
    const float* __restrict__ A, const float* __restrict__ B,
    float* __restrict__ C, int M, int Nc, int ldb) {
  constexpr int ldbs = NT * 16;
  __shared__ float Bs[(K / 2) * (2 * ldbs)];
  const int tid = threadIdx.x;
  for (int i = tid; i < K * ldbs; i += 256) {
    int k = i / ldbs, n = i - k * ldbs;
    float v = (n < Nc) ? B[k * ldb + n] : 0.f;
    Bs[(k >> 1) * (2 * ldbs) + 2 * n + (k & 1)] = v;
  }
  __syncthreads();

  const int wave = tid >> 5, lane = tid & 31;
  const int m0 = (blockIdx.x * 8 + wave) << 4;
  if (m0 >= M) return;                        // wave-uniform: EXEC stays full

  const int kb  = (lane >> 4) << 1;
  const int ncl = lane & 15;
  const float* Arow = A + (size_t)(m0 + (lane & 15)) * K;
  const float* Bp   = &Bs[2 * ncl];

  v8f acc[NT];
#pragma unroll
  for (int t = 0; t < NT; ++t)
#pragma unroll
    for (int r = 0; r < 8; ++r) acc[t][r] = 0.f;

#pragma unroll
  for (int k = 0; k < K; k += 4) {
    v2f a = *(const v2f*)(Arow + k + kb);
    const float* bpk = Bp + ((k + kb) >> 1) * (2 * ldbs);
#pragma unroll
    for (int t = 0; t < NT; ++t) {
      v2f b = *(const v2f*)(bpk + t * 32);
      acc[t] = __builtin_amdgcn_wmma_f32_16x16x4_f32(
          false, a, false, b, (short)0, acc[t], false, false);
    }
  }

  const int rbase = m0 + ((lane >> 4) << 3);
#pragma unroll
  for (int t = 0; t < NT; ++t) {
    int col = (t << 4) + ncl;
    if (col < Nc)
#pragma unroll
      for (int r = 0; r < 8; ++r)
        C[(size_t)(rbase + r) * Nc + col] = acc[t][r];
  }
}

// ---------------------------------------------------------------------------
// per-(node,head) attention coefficients: as = <h[n,h,:], a_s[h,:]>
// ---------------------------------------------------------------------------
template <int H, int C>
__global__ void k_node_alpha(const float* __restrict__ h,
                             const float* __restrict__ a_s,
                             const float* __restrict__ a_d,
                             float* __restrict__ as_, float* __restrict__ ad_,
                             int Nn) {
  int idx = blockIdx.x * blockDim.x + threadIdx.x;
  if (idx >= Nn * H) return;
  int n = idx / H, hh = idx - (idx / H) * H;
  const float* hp = h + (size_t)n * (H * C) + hh * C;
  float s1 = 0.f, s2 = 0.f;
#pragma unroll 8
  for (int c = 0; c < C; ++c) {
    float v = hp[c];
    s1 += v * a_s[hh * C + c];
    s2 += v * a_d[hh * C + c];
  }
  as_[idx] = s1;
  ad_[idx] = s2;
}

// edge pass 1: logits + segment max ----------------------------------------
template <int H>
__global__ void k_edge_logit_max(const int* __restrict__ srcA,
                                 const int* __restrict__ dstA, int Ne,
                                 const float* __restrict__ as_,
                                 const float* __restrict__ ad_,
                                 float* __restrict__ ebuf,
                                 float* __restrict__ mbuf, unsigned tot) {
  unsigned idx = blockIdx.x * blockDim.x + threadIdx.x;
  if (idx >= tot) return;
  int e = idx / H, hh = idx - (idx / H) * H;
  int s = (e < Ne) ? srcA[e] : (e - Ne);
  int d = (e < Ne) ? dstA[e] : (e - Ne);
  float v = as_[s * H + hh] + ad_[d * H + hh];
  v = (v > 0.f) ? v : NEG_SLOPE * v;
  ebuf[idx] = v;
  atomicMaxF(&mbuf[d * H + hh], v);
}

// edge pass 2: exp(e - m[dst]) + segment sum -------------------------------
template <int H>
__global__ void k_edge_exp_sum(const int* __restrict__ dstA, int Ne,
                               float* __restrict__ ebuf,
                               const float* __restrict__ mbuf,
                               float* __restrict__ sbuf, unsigned tot) {
  unsigned idx = blockIdx.x * blockDim.x + threadIdx.x;
  if (idx >= tot) return;
  int e = idx / H, hh = idx - (idx / H) * H;
  int d = (e < Ne) ? dstA[e] : (e - Ne);
  float v = __expf(ebuf[idx] - mbuf[d * H + hh]);
  ebuf[idx] = v;
  atomicAdd(&sbuf[d * H + hh], v);
}

// edge pass 3: weighted scatter-add, 4 channels / thread (b128 gather) ------
template <int H, int C>
__global__ void k_edge_agg(const int* __restrict__ srcA,
                           const int* __restrict__ dstA, int Ne,
                           const float* __restrict__ hbuf,
                           const float* __restrict__ ebuf,
                           const float* __restrict__ sbuf,
                           float* __restrict__ agg, unsigned tot) {
  constexpr int HC = H * C;       // 64 or 40, both divisible by 4
  constexpr int QV = HC / 4;
  unsigned idx = blockIdx.x * blockDim.x + threadIdx.x;
  if (idx >= tot) return;
  unsigned e = idx / QV;
  int q = (int)(idx - e * QV);
  int j = q * 4;
  int hh = j / C;
  int s = (e < (unsigned)Ne) ? srcA[e] : (int)(e - Ne);
  int d = (e < (unsigned)Ne) ? dstA[e] : (int)(e - Ne);
  float alpha = ebuf[e * H + hh] / (sbuf[d * H + hh] + 1e-16f);
  const float4 hv = *(const float4*)(hbuf + (size_t)s * HC + j);
  float* ap = agg + (size_t)d * HC + j;
  atomicAdd(ap + 0, hv.x * alpha);
  atomicAdd(ap + 1, hv.y * alpha);
  atomicAdd(ap + 2, hv.z * alpha);
  atomicAdd(ap + 3, hv.w * alpha);
}

// small helpers -------------------------------------------------------------
__global__ void k_fill(float* __restrict__ p, unsigned n, float v) {
  unsigned i = blockIdx.x * blockDim.x + threadIdx.x;
  if (i < n) p[i] = v;
}

// agg[i] = bias[i % HC], vectorized x4 (HC % 4 == 0)
__global__ void k_init_agg_bias4(float* __restrict__ agg,
                                 const float* __restrict__ bias, unsigned tot4,
                                 int HC) {
  unsigned i = blockIdx.x * blockDim.x + threadIdx.x;
  if (i >= tot4) return;
  unsigned base = i * 4u;
  float4 bv = *(const float4*)(bias + (base % (unsigned)HC));
  *(float4*)(agg + base) = bv;
}

// BatchNorm stats: sum & sumsq per channel (64 channels), LDS pre-reduce ----
__global__ __launch_bounds__(256) void k_bn_stats(const float* __restrict__ x,
                                                  float* __restrict__ stats,
                                                  int Nn) {
  __shared__ float ls[128];
  int tid = threadIdx.x;
  if (tid < 128) ls[tid] = 0.f;
  __syncthreads();
  unsigned total = (unsigned)Nn * 64u;
  for (unsigned i = blockIdx.x * blockDim.x + tid; i < total;
       i += gridDim.x * blockDim.x) {
    int c = i & 63;
    float v = x[i];
    atomicAdd(&ls[c], v);
    atomicAdd(&ls[64 + c], v * v);
  }
  __syncthreads();
  if (tid < 128) atomicAdd(&stats[tid], ls[tid]);
}

// y = relu(bn(x)), vectorized x4 over the 64-channel rows -------------------
__global__ void k_bn_relu4(const float* __restrict__ x,
                           const float* __restrict__ stats,
                           const float* __restrict__ g,
                           const float* __restrict__ be, float* __restrict__ y,
                           int Nn) {
  unsigned i = blockIdx.x * blockDim.x + threadIdx.x;
  unsigned tot = (unsigned)Nn * 16u;  // (Nn*64)/4
  if (i >= tot) return;
  unsigned base = i * 4u;
  int c0 = (int)(base & 63u);
  float invN = 1.0f / (float)Nn;
  float4 xv = *(const float4*)(x + base);
  float o[4] = {xv.x, xv.y, xv.z, xv.w};
#pragma unroll
  for (int r = 0; r < 4; ++r) {
    int c = c0 + r;
    float mu = stats[c] * invN;
    float var = stats[64 + c] * invN - mu * mu;
    float rs = rsqrtf(var + 1e-5f);
    float v = (o[r] - mu) * rs * g[c] + be[c];
    o[r] = v > 0.f ? v : 0.f;
  }
  *(float4*)(y + base) = make_float4(o[0], o[1], o[2], o[3]);
}

__global__ void k_log_softmax(const float* __restrict__ x,
                              float* __restrict__ out, int Nn, int Cc) {
  int n = blockIdx.x * blockDim.x + threadIdx.x;
  if (n >= Nn) return;
  const float* r = x + (size_t)n * Cc;
  float mx = -INFINITY;
  for (int c = 0; c < Cc; ++c) mx = fmaxf(mx, r[c]);
  float s = 0.f;
  for (int c = 0; c < Cc; ++c) s += __expf(r[c] - mx);
  float ls = __logf(s);
  for (int c = 0; c < Cc; ++c) out[(size_t)n * Cc + c] = r[c] - mx - ls;
}

// ---------------------------------------------------------------------------
static inline unsigned cdiv(unsigned long long a, unsigned long long b) {
  return (unsigned)((a + b - 1) / b);
}

extern "C" void kernel_launch(void* const* d_in, const int* in_sizes, int n_in,
                              void* d_out, int out_size, void* d_ws,
                              size_t ws_size, hipStream_t stream) {
  const float* x   = (const float*)d_in[0];
  const int*   ei  = (const int*)d_in[1];
  const float* W0  = (const float*)d_in[2];
  const float* aS0 = (const float*)d_in[3];
  const float* aD0 = (const float*)d_in[4];
  const float* b0  = (const float*)d_in[5];
  const float* g0  = (const float*)d_in[6];
  const float* be0 = (const float*)d_in[7];
  const float* W1  = (const float*)d_in[8];
  const float* aS1 = (const float*)d_in[9];
  const float* aD1 = (const float*)d_in[10];
  const float* b1  = (const float*)d_in[11];
  const float* g1  = (const float*)d_in[12];
  const float* be1 = (const float*)d_in[13];
  const float* W2  = (const float*)d_in[14];
  const float* aS2 = (const float*)d_in[15];
  const float* aD2 = (const float*)d_in[16];
  const float* b2  = (const float*)d_in[17];

  const int Nn = in_sizes[0] / 128;
  const int Ne = in_sizes[1] / 2;
  const int Et = Ne + Nn;
  const int* srcA = ei;
  const int* dstA = ei + Ne;

  // workspace layout (floats)
  float* ws = (float*)d_ws;
  size_t o = 0;
  float* hbuf  = ws + o; o += (size_t)Nn * 64;
  float* xbuf  = ws + o; o += (size_t)Nn * 64;
  float* agg   = ws + o; o += (size_t)Nn * 64;
  float* asb   = ws + o; o += (size_t)Nn * 4;
  float* adb   = ws + o; o += (size_t)Nn * 4;
  float* mbuf  = ws + o; o += (size_t)Nn * 4;
  float* sbuf  = ws + o; o += (size_t)Nn * 4;
  float* stats = ws + o; o += 128;
  float* ebuf  = ws + o; o += (size_t)Et * 4;

  const unsigned mtiles = (unsigned)((Nn + 15) / 16);
  const unsigned gemmBlocks = cdiv(mtiles, 8);

  // ===================== layer 0: 128 -> 4x16 ============================
  k_wmma_gemm<4, 128><<<gemmBlocks, 256, 0, stream>>>(x, W0, hbuf, Nn, 64, 64);
  k_node_alpha<4, 16><<<cdiv((size_t)Nn * 4, 256), 256, 0, stream>>>(
      hbuf, aS0, aD0, asb, adb, Nn);
  k_fill<<<cdiv((size_t)Nn * 4, 256), 256, 0, stream>>>(mbuf, Nn * 4, -INFINITY);
  k_fill<<<cdiv((size_t)Nn * 4, 256), 256, 0, stream>>>(sbuf, Nn * 4, 0.f);
  k_init_agg_bias4<<<cdiv((size_t)Nn * 16, 256), 256, 0, stream>>>(
      agg, b0, (unsigned)Nn * 16u, 64);
  {
    unsigned tot = (unsigned)Et * 4u;
    k_edge_logit_max<4><<<cdiv(tot, 256), 256, 0, stream>>>(
        srcA, dstA, Ne, asb, adb, ebuf, mbuf, tot);
    k_edge_exp_sum<4><<<cdiv(tot, 256), 256, 0, stream>>>(
        dstA, Ne, ebuf, mbuf, sbuf, tot);
    unsigned tota = (unsigned)Et * 16u;   // (edge, 4-ch group)
    k_edge_agg<4, 16><<<cdiv(tota, 256), 256, 0, stream>>>(
        srcA, dstA, Ne, hbuf, ebuf, sbuf, agg, tota);
  }
  k_fill<<<1, 256, 0, stream>>>(stats, 128, 0.f);
  k_bn_stats<<<1024, 256, 0, stream>>>(agg, stats, Nn);
  k_bn_relu4<<<cdiv((size_t)Nn * 16, 256), 256, 0, stream>>>(
      agg, stats, g0, be0, xbuf, Nn);

  // ===================== layer 1: 64 -> 4x16 =============================
  k_wmma_gemm<4, 64><<<gemmBlocks, 256, 0, stream>>>(xbuf, W1, hbuf, Nn, 64, 64);
  k_node_alpha<4, 16><<<cdiv((size_t)Nn * 4, 256), 256, 0, stream>>>(
      hbuf, aS1, aD1, asb, adb, Nn);
  k_fill<<<cdiv((size_t)Nn * 4, 256), 256, 0, stream>>>(mbuf, Nn * 4, -INFINITY);
  k_fill<<<cdiv((size_t)Nn * 4, 256), 256, 0, stream>>>(sbuf, Nn * 4, 0.f);
  k_init_agg_bias4<<<cdiv((size_t)Nn * 16, 256), 256, 0, stream>>>(
      agg, b1, (unsigned)Nn * 16u, 64);
  {
    unsigned tot = (unsigned)Et * 4u;
    k_edge_logit_max<4><<<cdiv(tot, 256), 256, 0, stream>>>(
        srcA, dstA, Ne, asb, adb, ebuf, mbuf, tot);
    k_edge_exp_sum<4><<<cdiv(tot, 256), 256, 0, stream>>>(
        dstA, Ne, ebuf, mbuf, sbuf, tot);
    unsigned tota = (unsigned)Et * 16u;
    k_edge_agg<4, 16><<<cdiv(tota, 256), 256, 0, stream>>>(
        srcA, dstA, Ne, hbuf, ebuf, sbuf, agg, tota);
  }
  k_fill<<<1, 256, 0, stream>>>(stats, 128, 0.f);
  k_bn_stats<<<1024, 256, 0, stream>>>(agg, stats, Nn);
  k_bn_relu4<<<cdiv((size_t)Nn * 16, 256), 256, 0, stream>>>(
      agg, stats, g1, be1, xbuf, Nn);

  // ===================== layer 2: 64 -> 1x40 =============================
  k_wmma_gemm<3, 64><<<gemmBlocks, 256, 0, stream>>>(xbuf, W2, hbuf, Nn, 40, 40);
  k_node_alpha<1, 40><<<cdiv((size_t)Nn, 256), 256, 0, stream>>>(
      hbuf, aS2, aD2, asb, adb, Nn);
  k_fill<<<cdiv((size_t)Nn, 256), 256, 0, stream>>>(mbuf, Nn, -INFINITY);
  k_fill<<<cdiv((size_t)Nn, 256), 256, 0, stream>>>(sbuf, Nn, 0.f);
  k_init_agg_bias4<<<cdiv((size_t)Nn * 10, 256), 256, 0, stream>>>(
      agg, b2, (unsigned)Nn * 10u, 40);
  {
    unsigned tot = (unsigned)Et;
    k_edge_logit_max<1><<<cdiv(tot, 256), 256, 0, stream>>>(
        srcA, dstA, Ne, asb, adb, ebuf, mbuf, tot);
    k_edge_exp_sum<1><<<cdiv(tot, 256), 256, 0, stream>>>(
        dstA, Ne, ebuf, mbuf, sbuf, tot);
    unsigned tota = (unsigned)Et * 10u;   // (edge, 4-ch group), HC=40
    k_edge_agg<1, 40><<<cdiv(tota, 256), 256, 0, stream>>>(
        srcA, dstA, Ne, hbuf, ebuf, sbuf, agg, tota);
  }
  k_log_softmax<<<cdiv((size_t)Nn, 256), 256, 0, stream>>>(
      agg, (float*)d_out, Nn, 40);
}